// DeepseekV4Indexer_39195871543898
// MI455X (gfx1250) — compile-verified
//
#include <hip/hip_runtime.h>
#include <hip/hip_bf16.h>

typedef float v2f __attribute__((ext_vector_type(2)));
typedef float v8f __attribute__((ext_vector_type(8)));

#define S_LEN   3072
#define HIDDEN  4096
#define NHEADS  64
#define HDIM    128
#define QDIM    (NHEADS * HDIM)   // 8192
#define NWIN    768
#define RATIO   4
#define TOPK_N  512
#define NEG_VAL (-1000000000.0f)
#define LOG_QT  9.210340371976184f    // ln(10000)
#define LOG_KT  11.982929094215963f   // ln(160000)

__device__ __forceinline__ v8f wmma_f32(v2f a, v2f b, v8f c) {
  // D = A(16x4) * B(4x16) + C, fp32 matrix pipe
  return __builtin_amdgcn_wmma_f32_16x16x4_f32(false, a, false, b, (short)0, c,
                                               false, false);
}

__device__ __forceinline__ v8f zero8() {
  v8f z = {0.f, 0.f, 0.f, 0.f, 0.f, 0.f, 0.f, 0.f};
  return z;
}

// CDNA5 async VMEM->LDS copies (ASYNCcnt-tracked, bypass VGPRs).
// LDS address is taken from the vdst VGPR; low 32 bits of a generic LDS pointer are
// exactly the wave-relative LDS byte offset on gfx1250.
__device__ __forceinline__ void async_load_b128(void* lds_ptr,
                                                const float* gptr) {
  const uint32_t lds_off = (uint32_t)(uintptr_t)lds_ptr;
  asm volatile("global_load_async_to_lds_b128 %0, %1, off"
               :
               : "v"(lds_off), "v"(gptr)
               : "memory");
}

__device__ __forceinline__ void async_load_b32(void* lds_ptr,
                                               const float* gptr) {
  const uint32_t lds_off = (uint32_t)(uintptr_t)lds_ptr;
  asm volatile("global_load_async_to_lds_b32 %0, %1, off"
               :
               : "v"(lds_off), "v"(gptr)
               : "memory");
}

__device__ __forceinline__ void wait_async0() {
  asm volatile("s_wait_asynccnt 0" ::: "memory");
}

// ---------------------------------------------------------------------------
// Generic C[3072 x N] = A[3072 x K] * B[K x N], f32, WMMA 16x16x4.
// Block: 256 thr (8 waves), tile 64 x 128; wave = 16 x 64 (4 accumulators).
// Double-buffered async-to-LDS staging overlapped with WMMA compute.
// Both As and Bs are [major][k] with stride 20 (pad) -> fragment reads are
// contiguous b64 pairs, bank-conflict-free, no register repacking.
// ---------------------------------------------------------------------------
__global__ __launch_bounds__(256) void gemm_f32_wmma(
    const float* __restrict__ A, const float* __restrict__ B,
    float* __restrict__ C, int N, int K) {
  __shared__ float As[2][64 * 20];   // [row][k]
  __shared__ float Bs[2][128 * 20];  // [col][k]

  const int tid  = threadIdx.x;
  const int lane = tid & 31;
  const int wave = tid >> 5;
  const int wr   = wave >> 1;       // 0..3 -> row slice
  const int wc   = wave & 1;        // 0..1 -> col slice
  const int l16  = lane & 15;
  const int hi   = lane >> 4;       // half-wave selects K pair {0,1} vs {2,3}
  const int kb   = hi * 2;
  const int row0 = blockIdx.x * 64;
  const int col0 = blockIdx.y * 128;

  // staging assignments
  const int arow = tid >> 2;        // 0..63
  const int akq  = (tid & 3) * 4;   // 0,4,8,12
  // B: 2048 elements, 8 per thread; element e -> k row (e>>7), col (e&127)
  int bcol[8];
  #pragma unroll
  for (int i = 0; i < 8; ++i) {
    const int e = tid + 256 * i;
    int col = col0 + (e & 127);
    if (col > N - 1) col = N - 1;   // OOB cols read valid garbage; store-masked
    bcol[i] = col;
  }

  v8f acc[4];
  #pragma unroll
  for (int t = 0; t < 4; ++t) acc[t] = zero8();

  // prologue: stage tile 0
  {
    async_load_b128(&As[0][arow * 20 + akq],
                    A + (size_t)(row0 + arow) * K + akq);
    #pragma unroll
    for (int i = 0; i < 8; ++i) {
      const int e = tid + 256 * i;
      const int j = e >> 7, c = e & 127;
      async_load_b32(&Bs[0][c * 20 + j], B + (size_t)j * N + bcol[i]);
    }
  }
  wait_async0();
  __syncthreads();

  const int nIter = K / 16;
  for (int it = 0; it < nIter; ++it) {
    const int cur = it & 1;
    const bool more = (it + 1) < nIter;
    if (more) {  // stage next tile into the other buffer (async, no wait yet)
      const int k0 = (it + 1) * 16;
      async_load_b128(&As[1 - cur][arow * 20 + akq],
                      A + (size_t)(row0 + arow) * K + k0 + akq);
      #pragma unroll
      for (int i = 0; i < 8; ++i) {
        const int e = tid + 256 * i;
        const int j = e >> 7, c = e & 127;
        async_load_b32(&Bs[1 - cur][c * 20 + j],
                       B + (size_t)(k0 + j) * N + bcol[i]);
      }
    }
    // compute on current tile (overlaps with the async copies above)
    #pragma unroll
    for (int k4 = 0; k4 < 4; ++k4) {
      const int koff = k4 * 4 + kb;
      const v2f a = *(const v2f*)(&As[cur][(wr * 16 + l16) * 20 + koff]);
      #pragma unroll
      for (int t = 0; t < 4; ++t) {
        const int cc = wc * 64 + t * 16 + l16;
        const v2f b = *(const v2f*)(&Bs[cur][cc * 20 + koff]);
        acc[t] = wmma_f32(a, b, acc[t]);
      }
    }
    if (more) wait_async0();
    __syncthreads();
  }

  // D layout: VGPR r -> row r + 8*hi, lane%16 -> col
  #pragma unroll
  for (int t = 0; t < 4; ++t) {
    const int c = col0 + wc * 64 + t * 16 + l16;
    if (c < N) {
      #pragma unroll
      for (int r = 0; r < 8; ++r) {
        const int m = row0 + wr * 16 + r + hi * 8;
        C[(size_t)m * N + c] = acc[t][r];
      }
    }
  }
}

// ---------------------------------------------------------------------------
// Interleaved RoPE on q (layout [s][h*128+d]), in place. One thread per pair.
// ---------------------------------------------------------------------------
__global__ __launch_bounds__(256) void rope_q_kernel(float* __restrict__ q) {
  const int idx = blockIdx.x * 256 + threadIdx.x;  // S*NHEADS*64
  const int p = idx & 63;
  const int h = (idx >> 6) & 63;
  const int s = idx >> 12;
  const float inv = __expf(-(float)p * (1.0f / 64.0f) * LOG_QT);
  float sn, cs;
  __sincosf((float)s * inv, &sn, &cs);
  float* ptr = q + (size_t)s * QDIM + h * HDIM + 2 * p;
  const float2 x = *(const float2*)ptr;
  *(float2*)ptr = make_float2(x.x * cs - x.y * sn, x.y * cs + x.x * sn);
}

// ---------------------------------------------------------------------------
// Overlap pool + gate softmax (8-way) + RoPE(theta_k) -> k_c[768][128].
// One thread per (window, pair).
// ---------------------------------------------------------------------------
__global__ __launch_bounds__(256) void pool_kc_kernel(
    const float* __restrict__ kv, const float* __restrict__ gate,
    float* __restrict__ kc) {
  const int idx = blockIdx.x * 256 + threadIdx.x;  // NWIN*64
  const int p = idx & 63;
  const int w = idx >> 6;
  float out[2];
  #pragma unroll
  for (int e = 0; e < 2; ++e) {
    const int d = 2 * p + e;
    float g[8], v[8];
    #pragma unroll
    for (int r = 0; r < 4; ++r) {
      const int si = w * RATIO + r;                 // hi half (this window)
      g[4 + r] = gate[(size_t)si * 256 + 128 + d];
      v[4 + r] = kv[(size_t)si * 256 + 128 + d];
      if (w > 0) {                                  // lo half (prev window)
        const int sj = (w - 1) * RATIO + r;
        g[r] = gate[(size_t)sj * 256 + d];
        v[r] = kv[(size_t)sj * 256 + d];
      } else {
        g[r] = -3.0e38f;
        v[r] = 0.f;
      }
    }
    float m = g[0];
    #pragma unroll
    for (int r = 1; r < 8; ++r) m = fmaxf(m, g[r]);
    float den = 0.f, num = 0.f;
    #pragma unroll
    for (int r = 0; r < 8; ++r) {
      const float ex = __expf(g[r] - m);
      den += ex;
      num += ex * v[r];
    }
    out[e] = num / den;
  }
  const float inv = __expf(-(float)p * (1.0f / 64.0f) * LOG_KT);
  float sn, cs;
  __sincosf((float)(w * RATIO) * inv, &sn, &cs);
  kc[(size_t)w * HDIM + 2 * p]     = out[0] * cs - out[1] * sn;
  kc[(size_t)w * HDIM + 2 * p + 1] = out[1] * cs + out[0] * sn;
}

// ---------------------------------------------------------------------------
// Fused: I[s,w] = mask( sum_h wmix[s,h] * relu(q[h,s,:]·k_c[w,:]) ).
// Block: 128 thr (4 waves) -> tile 16(s) x 64(w). Waves split the 64 heads;
// per head-pass: 32 k-steps x 4 WMMAs; A frags streamed from global (L2-
// resident q), B frags from async-staged LDS K-tile. ds_add_f32 reduction.
// ---------------------------------------------------------------------------
__global__ __launch_bounds__(128) void scores_i_kernel(
    const float* __restrict__ q, const float* __restrict__ kc,
    const float* __restrict__ wmix, float* __restrict__ I) {
  __shared__ float Ks[64 * 132];   // [col][k], stride 132 (conflict-free b64)
  __shared__ float Wm[16 * 68];    // [row][h], stride 68 (conflict-free b32)
  __shared__ float Ps[16 * 68];    // partial I tile

  const int tid  = threadIdx.x;
  const int lane = tid & 31;
  const int wave = tid >> 5;       // 0..3
  const int l16  = lane & 15;
  const int hi   = lane >> 4;
  const int kb   = hi * 2;
  const int s0   = blockIdx.x * 16;
  const int w0   = blockIdx.y * 64;

  // Stage K tile 64x128 via async VMEM->LDS b128 copies
  for (int ci = tid; ci < 2048; ci += 128) {
    const int c  = ci >> 5;
    const int k4 = (ci & 31) * 4;
    async_load_b128(&Ks[c * 132 + k4],
                    kc + (size_t)(w0 + c) * HDIM + k4);
  }
  // Stage wmix tile 16x64 and zero the reduction tile
  for (int e = tid; e < 1024; e += 128) {
    const int i = e >> 6, h = e & 63;
    Wm[i * 68 + h] = wmix[(size_t)(s0 + i) * NHEADS + h];
    Ps[i * 68 + h] = 0.f;
  }
  wait_async0();
  __syncthreads();

  v8f acc[4];
  #pragma unroll
  for (int t = 0; t < 4; ++t) acc[t] = zero8();

  const float* qbase = q + (size_t)(s0 + l16) * QDIM;

  for (int h = wave; h < NHEADS; h += 4) {
    v8f sc[4];
    #pragma unroll
    for (int t = 0; t < 4; ++t) sc[t] = zero8();

    const float* qrow = qbase + h * HDIM + kb;
    #pragma unroll 4
    for (int k4 = 0; k4 < 32; ++k4) {
      const float2 a2 = *(const float2*)(qrow + k4 * 4);
      v2f a;
      a.x = a2.x;
      a.y = a2.y;
      #pragma unroll
      for (int t = 0; t < 4; ++t) {
        const v2f b = *(const v2f*)(&Ks[(t * 16 + l16) * 132 + k4 * 4 + kb]);
        sc[t] = wmma_f32(a, b, sc[t]);
      }
    }
    // relu -> scale by wmix[s,h] -> accumulate over heads
    #pragma unroll
    for (int r = 0; r < 8; ++r) {
      const float wm = Wm[(r + hi * 8) * 68 + h];
      #pragma unroll
      for (int t = 0; t < 4; ++t) {
        float v = sc[t][r];
        v = v > 0.f ? v : 0.f;
        acc[t][r] += v * wm;
      }
    }
  }

  // Cross-wave head reduction in LDS
  #pragma unroll
  for (int t = 0; t < 4; ++t) {
    #pragma unroll
    for (int r = 0; r < 8; ++r)
      atomicAdd(&Ps[(r + hi * 8) * 68 + (t * 16 + l16)], acc[t][r]);
  }
  __syncthreads();

  // Mask + store
  for (int e = tid; e < 1024; e += 128) {
    const int m = e >> 6, c = e & 63;
    const int s = s0 + m, w = w0 + c;
    float val = Ps[m * 68 + c];
    if (w * RATIO + RATIO - 1 > s) val = NEG_VAL;
    I[(size_t)s * NWIN + w] = val;
  }
}

// ---------------------------------------------------------------------------
// Exact stable top-512 of 768 per row by rank counting (matches jax top_k
// tie order). One block per row; values broadcast-read from LDS.
// ---------------------------------------------------------------------------
__global__ __launch_bounds__(256) void topk_kernel(
    const float* __restrict__ I, float* __restrict__ topk_idx) {
  __shared__ float vals[NWIN];
  const int s = blockIdx.x;
  const int tid = threadIdx.x;
  for (int e = tid; e < NWIN; e += 256) vals[e] = I[(size_t)s * NWIN + e];
  __syncthreads();
  for (int e = tid; e < NWIN; e += 256) {
    const float v = vals[e];
    int rank = 0;
    for (int j = 0; j < NWIN; ++j) {
      const float u = vals[j];
      rank += (u > v) || (u == v && j < e);
    }
    if (rank < TOPK_N) topk_idx[(size_t)s * TOPK_N + rank] = (float)e;
  }
}

// ---------------------------------------------------------------------------
extern "C" void kernel_launch(void* const* d_in, const int* in_sizes, int n_in,
                              void* d_out, int out_size, void* d_ws,
                              size_t ws_size, hipStream_t stream) {
  const float* hidden = (const float*)d_in[0];
  const float* wq     = (const float*)d_in[1];
  const float* wk     = (const float*)d_in[2];
  const float* wg     = (const float*)d_in[3];
  const float* ww     = (const float*)d_in[4];

  float* I_out    = (float*)d_out;                       // [3072][768]
  float* topk_out = I_out + (size_t)S_LEN * NWIN;        // [3072][512]

  float* q    = (float*)d_ws;                            // [3072][8192]
  float* kv   = q + (size_t)S_LEN * QDIM;                // [3072][256]
  float* gate = kv + (size_t)S_LEN * 256;                // [3072][256]
  float* wmix = gate + (size_t)S_LEN * 256;              // [3072][64]
  float* kc   = wmix + (size_t)S_LEN * NHEADS;           // [768][128]

  gemm_f32_wmma<<<dim3(S_LEN / 64, QDIM / 128), 256, 0, stream>>>(
      hidden, wq, q, QDIM, HIDDEN);
  gemm_f32_wmma<<<dim3(S_LEN / 64, 2), 256, 0, stream>>>(
      hidden, wk, kv, 256, HIDDEN);
  gemm_f32_wmma<<<dim3(S_LEN / 64, 2), 256, 0, stream>>>(
      hidden, wg, gate, 256, HIDDEN);
  gemm_f32_wmma<<<dim3(S_LEN / 64, 1), 256, 0, stream>>>(
      hidden, ww, wmix, NHEADS, HIDDEN);

  rope_q_kernel<<<(S_LEN * NHEADS * 64) / 256, 256, 0, stream>>>(q);
  pool_kc_kernel<<<(NWIN * 64) / 256, 256, 0, stream>>>(kv, gate, kc);

  scores_i_kernel<<<dim3(S_LEN / 16, NWIN / 64), 128, 0, stream>>>(
      q, kc, wmix, I_out);
  topk_kernel<<<S_LEN, 256, 0, stream>>>(I_out, topk_out);
}